// GenePredictionT5_52398601011638
// MI455X (gfx1250) — compile-verified
//
#include <hip/hip_runtime.h>
#include <hip/hip_bf16.h>
#include <math.h>

typedef float v2f __attribute__((ext_vector_type(2)));
typedef float v8f __attribute__((ext_vector_type(8)));

#define NB 2
#define MM 256
#define DM 256

__device__ __forceinline__ float gelu_f(float x) {
  return 0.5f * x * (1.0f + erff(x * 0.7071067811865476f));
}

// ---------------------------------------------------------------------------
// Implicit-im2col convolution (NCHW, 1 x KW kernels) via V_WMMA_F32_16X16X4_F32.
// 64x64 block tile, 4 waves, each wave owns a 2x2 grid of 16x16 f32 accums.
// Optional fused input activation  gelu(x * g/sqrt(1+eps) + b)  (per channel),
// optional residual add (same NCHW layout as output).
// rows = NB*MM*Wout (multiple of 64 here), cols = OC (multiple of 64).
// ---------------------------------------------------------------------------
__global__ __launch_bounds__(128) void conv_wmma(
    const float* __restrict__ X, const float* __restrict__ Wt,
    const float* __restrict__ Res, float* __restrict__ Out,
    const float* __restrict__ bnG, const float* __restrict__ bnB, int actOn,
    int Cin, int Win, int Wout, int KW, int pad, int OC)
{
  const int K = Cin * KW;
  const int planeMW = MM * Wout;
  const int rows = NB * planeMW;
  __shared__ float As[64][16];
  __shared__ float Bs[16][64];
  const int tid = threadIdx.x;
  const int lane = tid & 31, wid = tid >> 5;
  const int tileR = blockIdx.y * 64, tileC = blockIdx.x * 64;
  v8f acc[2][2] = {};
  const float kInvS = 0.9999950000374997f; // 1/sqrt(1+1e-5)

  for (int k0 = 0; k0 < K; k0 += 16) {
    // A tile: 64 rows x 16 k, implicit im2col with fused bn+gelu
    for (int e = tid; e < 1024; e += 128) {
      int r = e >> 4, k = e & 15;
      int gr = tileR + r, gk = k0 + k;
      float v = 0.f;
      if (gr < rows && gk < K) {
        int n = gr / planeMW; int rem = gr - n * planeMW;
        int m = rem / Wout;   int w = rem - m * Wout;
        int ci = gk / KW;     int kk = gk - ci * KW;
        int iw = w + kk - pad;
        if (iw >= 0 && iw < Win) {
          float xv = X[(((long)n * Cin + ci) * MM + m) * (long)Win + iw];
          if (actOn) {
            xv = xv * (bnG[ci] * kInvS) + bnB[ci];
            xv = gelu_f(xv);
          }
          v = xv;
        }
      }
      As[r][k] = v;
    }
    // B tile: weights (OC, Cin*KW) row-major -> Bs[k][oc]
    for (int e = tid; e < 1024; e += 128) {
      int k = e >> 6, c = e & 63;
      int gk = k0 + k, gc = tileC + c;
      Bs[k][c] = (gk < K && gc < OC) ? Wt[(long)gc * K + gk] : 0.f;
      if (gk + 16 < K && gc < OC)
        __builtin_prefetch(&Wt[(long)gc * K + gk + 16], 0, 3);
    }
    __syncthreads();
    {
      const int wr = (wid >> 1) << 5, wc = (wid & 1) << 5;
      const int l16 = lane & 15, kh = (lane >> 4) << 1;
#pragma unroll
      for (int kk = 0; kk < 16; kk += 4) {
        v2f a0, a1, b0, b1;
        a0.x = As[wr + l16][kk + kh];
        a0.y = As[wr + l16][kk + kh + 1];
        a1.x = As[wr + 16 + l16][kk + kh];
        a1.y = As[wr + 16 + l16][kk + kh + 1];
        b0.x = Bs[kk + kh][wc + l16];
        b0.y = Bs[kk + kh + 1][wc + l16];
        b1.x = Bs[kk + kh][wc + 16 + l16];
        b1.y = Bs[kk + kh + 1][wc + 16 + l16];
        acc[0][0] = __builtin_amdgcn_wmma_f32_16x16x4_f32(false, a0, false, b0,
                                                          (short)0, acc[0][0], false, false);
        acc[0][1] = __builtin_amdgcn_wmma_f32_16x16x4_f32(false, a0, false, b1,
                                                          (short)0, acc[0][1], false, false);
        acc[1][0] = __builtin_amdgcn_wmma_f32_16x16x4_f32(false, a1, false, b0,
                                                          (short)0, acc[1][0], false, false);
        acc[1][1] = __builtin_amdgcn_wmma_f32_16x16x4_f32(false, a1, false, b1,
                                                          (short)0, acc[1][1], false, false);
      }
    }
    __syncthreads();
  }

  const int wr = (wid >> 1) << 5, wc = (wid & 1) << 5;
  const int rhalf = (lane >> 4) << 3;
#pragma unroll
  for (int i = 0; i < 2; ++i) {
#pragma unroll
    for (int j = 0; j < 2; ++j) {
      int gcol = tileC + wc + 16 * j + (lane & 15);
#pragma unroll
      for (int r = 0; r < 8; ++r) {
        int grow = tileR + wr + 16 * i + r + rhalf;
        if (grow < rows && gcol < OC) {
          int n = grow / planeMW; int rem = grow - n * planeMW;
          long o = (long)n * OC * planeMW + (long)gcol * planeMW + rem;
          float v = acc[i][j][r];
          if (Res) v += Res[o];
          Out[o] = v;
        }
      }
    }
  }
}

// ---------------------------------------------------------------------------
// Generic strided WMMA GEMM with 2-level batch strides (z = z1*Z0 + z0):
//   out = op(alpha * (A@B) + R1) + R2,  op in {identity, gelu}
// A[i][k] = A[z0*a_b0+z1*a_b1 + i*a_rs + k*a_cs], same scheme for B/C/R1/R2.
// 64x64 block tile, 2x2 accumulators per wave.
// ---------------------------------------------------------------------------
__global__ __launch_bounds__(128) void gemm_wmma(
    const float* __restrict__ A, long a_rs, long a_cs, long a_b0, long a_b1,
    const float* __restrict__ B, long b_rs, long b_cs, long b_b0, long b_b1,
    float* __restrict__ C, long c_rs, long c_cs, long c_b0, long c_b1,
    const float* __restrict__ R1, long r1_rs, long r1_cs, long r1_b0, long r1_b1,
    const float* __restrict__ R2, long r2_rs, long r2_cs, long r2_b0, long r2_b1,
    int rows, int cols, int K, int Z0, float alpha, int opGelu)
{
  __shared__ float As[64][16];
  __shared__ float Bs[16][64];
  const int z = blockIdx.z, z0 = z % Z0, z1 = z / Z0;
  const float* Ab = A + z0 * a_b0 + z1 * a_b1;
  const float* Bb = B + z0 * b_b0 + z1 * b_b1;
  float* Cb = C + z0 * c_b0 + z1 * c_b1;
  const int tid = threadIdx.x, lane = tid & 31, wid = tid >> 5;
  const int tileR = blockIdx.y * 64, tileC = blockIdx.x * 64;
  v8f acc[2][2] = {};

  for (int k0 = 0; k0 < K; k0 += 16) {
    for (int e = tid; e < 1024; e += 128) {
      int r = e >> 4, k = e & 15;
      int gr = tileR + r, gk = k0 + k;
      As[r][k] = (gr < rows && gk < K) ? Ab[(long)gr * a_rs + (long)gk * a_cs] : 0.f;
      if (gr < rows && gk + 16 < K)
        __builtin_prefetch(&Ab[(long)gr * a_rs + (long)(gk + 16) * a_cs], 0, 3);
    }
    for (int e = tid; e < 1024; e += 128) {
      int k = e >> 6, c = e & 63;
      int gk = k0 + k, gc = tileC + c;
      Bs[k][c] = (gk < K && gc < cols) ? Bb[(long)gk * b_rs + (long)gc * b_cs] : 0.f;
      if (gk + 16 < K && gc < cols)
        __builtin_prefetch(&Bb[(long)(gk + 16) * b_rs + (long)gc * b_cs], 0, 3);
    }
    __syncthreads();
    {
      const int wr = (wid >> 1) << 5, wc = (wid & 1) << 5;
      const int l16 = lane & 15, kh = (lane >> 4) << 1;
#pragma unroll
      for (int kk = 0; kk < 16; kk += 4) {
        v2f a0, a1, b0, b1;
        a0.x = As[wr + l16][kk + kh];
        a0.y = As[wr + l16][kk + kh + 1];
        a1.x = As[wr + 16 + l16][kk + kh];
        a1.y = As[wr + 16 + l16][kk + kh + 1];
        b0.x = Bs[kk + kh][wc + l16];
        b0.y = Bs[kk + kh + 1][wc + l16];
        b1.x = Bs[kk + kh][wc + 16 + l16];
        b1.y = Bs[kk + kh + 1][wc + 16 + l16];
        acc[0][0] = __builtin_amdgcn_wmma_f32_16x16x4_f32(false, a0, false, b0,
                                                          (short)0, acc[0][0], false, false);
        acc[0][1] = __builtin_amdgcn_wmma_f32_16x16x4_f32(false, a0, false, b1,
                                                          (short)0, acc[0][1], false, false);
        acc[1][0] = __builtin_amdgcn_wmma_f32_16x16x4_f32(false, a1, false, b0,
                                                          (short)0, acc[1][0], false, false);
        acc[1][1] = __builtin_amdgcn_wmma_f32_16x16x4_f32(false, a1, false, b1,
                                                          (short)0, acc[1][1], false, false);
      }
    }
    __syncthreads();
  }

  const int wr = (wid >> 1) << 5, wc = (wid & 1) << 5;
  const int rhalf = (lane >> 4) << 3;
#pragma unroll
  for (int i = 0; i < 2; ++i) {
#pragma unroll
    for (int j = 0; j < 2; ++j) {
      int gcol = tileC + wc + 16 * j + (lane & 15);
#pragma unroll
      for (int r = 0; r < 8; ++r) {
        int grow = tileR + wr + 16 * i + r + rhalf;
        if (grow < rows && gcol < cols) {
          float v = alpha * acc[i][j][r];
          if (R1) v += (R1 + z0 * r1_b0 + z1 * r1_b1)[(long)grow * r1_rs + (long)gcol * r1_cs];
          if (opGelu) v = gelu_f(v);
          if (R2) v += (R2 + z0 * r2_b0 + z1 * r2_b1)[(long)grow * r2_rs + (long)gcol * r2_cs];
          Cb[(long)grow * c_rs + (long)gcol * c_cs] = v;
        }
      }
    }
  }
}

// ---------------------------------------------------------------------------
// Max-pool along W (NCHW), writing into a concat buffer with Ctot channels at
// channel offset cOff.
// ---------------------------------------------------------------------------
__global__ void maxpool_k(const float* __restrict__ x, float* __restrict__ y,
                          int C, int Wout, int kwin, int Ctot, int cOff)
{
  long total = (long)NB * C * MM * Wout;
  long i = (long)blockIdx.x * blockDim.x + threadIdx.x;
  if (i >= total) return;
  int w = (int)(i % Wout); long r = i / Wout;
  int m = (int)(r % MM); r /= MM;
  int c = (int)(r % C); int n = (int)(r / C);
  const float* p = x + ((((long)n * C + c) * MM + m) * ((long)Wout * kwin)) + (long)w * kwin;
  float mv = p[0];
  for (int k = 1; k < kwin; ++k) mv = fmaxf(mv, p[k]);
  y[(((long)n * Ctot + cOff + c) * MM + m) * (long)Wout + w] = mv;
}

// (N,128,M,8) NCHW -> (N*M, 1024) tokens:  y[(n*M+m)*1024 + c*8+w]
__global__ void reshape_k(const float* __restrict__ x, float* __restrict__ y)
{
  int i = blockIdx.x * blockDim.x + threadIdx.x; // 524288 total
  int k = i & 1023; int row = i >> 10;
  int m = row & 255; int n = row >> 8;
  int c = k >> 3; int w = k & 7;
  y[i] = x[((((long)n * 128 + c) * 256 + m) << 3) + w];
}

// Relative-position bias expanded to (N*H, M, M), with -1e30 on the diagonal.
__global__ void rpbias_k(const float* __restrict__ emb, float* __restrict__ out)
{
  int idx = blockIdx.x * blockDim.x + threadIdx.x; // 8*256*256
  int j = idx & 255; int i = (idx >> 8) & 255; int z = idx >> 16; int h = z & 3;
  int rp = i - j;
  int ret = (rp < 0) ? 32 : 0;
  int arp = (rp < 0) ? -rp : rp;
  int bucket;
  if (arp < 16) {
    bucket = ret + arp;
  } else {
    int vl = 16 + (int)(logf((float)arp * (1.0f / 16.0f)) * (16.0f / logf(16.0f)));
    if (vl > 31) vl = 31;
    bucket = ret + vl;
  }
  float v = emb[bucket * 4 + h];
  if (i == j) v += -1e30f;
  out[idx] = v;
}

__global__ void layernorm_k(const float* __restrict__ x, float* __restrict__ y,
                            const float* __restrict__ g, const float* __restrict__ b)
{
  __shared__ float red[256];
  int row = blockIdx.x, t = threadIdx.x;
  float v = x[(long)row * DM + t];
  red[t] = v; __syncthreads();
  for (int s = 128; s > 0; s >>= 1) { if (t < s) red[t] += red[t + s]; __syncthreads(); }
  float mu = red[0] * (1.0f / DM);
  __syncthreads();
  float d = v - mu;
  red[t] = d * d; __syncthreads();
  for (int s = 128; s > 0; s >>= 1) { if (t < s) red[t] += red[t + s]; __syncthreads(); }
  float var = red[0] * (1.0f / DM);
  y[(long)row * DM + t] = d * rsqrtf(var + 1e-5f) * g[t] + b[t];
}

__global__ void softmax_k(float* __restrict__ x)
{
  __shared__ float red[256];
  int row = blockIdx.x, t = threadIdx.x;
  float v = x[(long)row * 256 + t];
  red[t] = v; __syncthreads();
  for (int s = 128; s > 0; s >>= 1) { if (t < s) red[t] = fmaxf(red[t], red[t + s]); __syncthreads(); }
  float mx = red[0]; __syncthreads();
  float e = expf(v - mx);
  red[t] = e; __syncthreads();
  for (int s = 128; s > 0; s >>= 1) { if (t < s) red[t] += red[t + s]; __syncthreads(); }
  x[(long)row * 256 + t] = e / red[0];
}

// Head: take token M/2, fc1(256->64)+gelu, fc2(64->1)+bias, softplus.
__global__ void head_k(const float* __restrict__ h,
                       const float* __restrict__ w1, const float* __restrict__ b1,
                       const float* __restrict__ w2, const float* __restrict__ b2,
                       float* __restrict__ out)
{
  __shared__ float mid[64];
  int n = blockIdx.x, t = threadIdx.x;
  const float* row = h + ((long)n * 256 + 128) * 256;
  float a = b1[t];
  for (int k = 0; k < 256; ++k) a += row[k] * w1[k * 64 + t];
  a = gelu_f(a);
  mid[t] = a * w2[t];
  __syncthreads();
  for (int s = 32; s > 0; s >>= 1) { if (t < s) mid[t] += mid[t + s]; __syncthreads(); }
  if (t == 0) {
    float z = mid[0] + b2[0];
    out[n] = (z > 20.f) ? z : log1pf(expf(z));
  }
}

// ---------------------------------------------------------------------------
extern "C" void kernel_launch(void* const* d_in, const int* in_sizes, int n_in,
                              void* d_out, int out_size, void* d_ws, size_t ws_size,
                              hipStream_t stream)
{
  (void)in_sizes; (void)n_in; (void)out_size; (void)ws_size;
  auto P = [&](int i) { return (const float*)d_in[i]; };
  float* ws = (float*)d_ws;

  const size_t BIG = 33554432;          // 2*128*256*512
  float* R0 = ws;
  float* R1 = ws + BIG;
  float* R2 = ws + 2 * BIG;             // concat / tower ping buffer (25165824)
  float* SMp = ws + 2 * BIG + 25165824;
  float* XT = SMp;                      // 512 x 1024
  float* Xb = XT + 524288;              // 512 x 256
  float* Hb = Xb + 131072;
  float* Qb = Hb + 131072;
  float* Kb = Qb + 131072;
  float* Vb = Kb + 131072;
  float* Sb = Vb + 131072;              // 8 x 256 x 256
  float* Ao = Sb + 524288;              // 512 x 256
  float* Fb = Ao + 131072;              // 512 x 1024
  float* Bx = Fb + 524288;              // 8 x 256 x 256 rp bias

  dim3 blk(128);

  // ---------------- stems ----------------
  struct StemD { const float* in; int cin; int wi; int ri; int coff; };
  StemD st[3] = { { P(0), 4, 3, 4, 0 },
                  { P(1), 1, 7, 8, 128 },
                  { P(2), 4, 11, 12, 256 } };
  for (int s = 0; s < 3; ++s) {
    dim3 g1(2, (NB * MM * 512) / 64);   // 2 x 4096
    // conv (OC,cin,1,30): 541 -> 512
    conv_wmma<<<g1, blk, 0, stream>>>(st[s].in, P(st[s].wi), nullptr, R0,
                                      nullptr, nullptr, 0,
                                      st[s].cin, 541, 512, 30, 0, 128);
    // rconv: R0 + conv1x1(gelu(bn(R0)))
    conv_wmma<<<g1, blk, 0, stream>>>(R0, P(st[s].ri + 2), R0, R1,
                                      P(st[s].ri), P(st[s].ri + 1), 1,
                                      128, 512, 512, 1, 0, 128);
    // maxpool4 into concat (2,384,256,128) at channel offset
    long tot = (long)NB * 128 * MM * 128;
    maxpool_k<<<dim3((unsigned)((tot + 255) / 256)), dim3(256), 0, stream>>>(
        R1, R2, 128, 128, 4, 384, st[s].coff);
  }

  // ---------------- tower ----------------
  int tbase[4] = { 15, 21, 27, 33 };
  int Cin = 384, Wc = 128;
  const float* cur = R2;
  for (int t = 0; t < 4; ++t) {
    int b = tbase[t];
    int kw = (t == 0) ? 7 : 3, pad = (t == 0) ? 3 : 1;
    dim3 g1(2, (NB * MM * Wc) / 64);
    conv_wmma<<<g1, blk, 0, stream>>>(cur, P(b + 2), nullptr, R0, P(b), P(b + 1), 1,
                                      Cin, Wc, Wc, kw, pad, 128);
    conv_wmma<<<g1, blk, 0, stream>>>(R0, P(b + 5), R0, R1, P(b + 3), P(b + 4), 1,
                                      128, Wc, Wc, 1, 0, 128);
    int Wo = Wc / 2;
    long tot = (long)NB * 128 * MM * Wo;
    maxpool_k<<<dim3((unsigned)((tot + 255) / 256)), dim3(256), 0, stream>>>(
        R1, R2, 128, Wo, 2, 128, 0);
    Cin = 128; Wc = Wo; cur = R2;
  }

  // ---------------- tokens + projection ----------------
  reshape_k<<<dim3(2048), dim3(256), 0, stream>>>(R2, XT);
  {
    dim3 gp(4, 8, 1);
    gemm_wmma<<<gp, blk, 0, stream>>>(
        XT, 1024, 1, 0, 0,
        P(39), 256, 1, 0, 0,
        Xb, 256, 1, 0, 0,
        P(40), 0, 1, 0, 0,
        nullptr, 0, 0, 0, 0,
        512, 256, 1024, 1, 1.0f, 0);
  }
  rpbias_k<<<dim3(2048), dim3(256), 0, stream>>>(P(41), Bx);

  // ---------------- transformer layers ----------------
  for (int l = 0; l < 4; ++l) {
    int b = 42 + 12 * l;
    layernorm_k<<<dim3(512), dim3(256), 0, stream>>>(Xb, Hb, P(b + 0), P(b + 1));
    dim3 gq(4, 8, 1);
    // Q, K, V
    gemm_wmma<<<gq, blk, 0, stream>>>(Hb, 256, 1, 0, 0, P(b + 2), 256, 1, 0, 0,
                                      Qb, 256, 1, 0, 0, nullptr, 0, 0, 0, 0,
                                      nullptr, 0, 0, 0, 0, 512, 256, 256, 1, 1.0f, 0);
    gemm_wmma<<<gq, blk, 0, stream>>>(Hb, 256, 1, 0, 0, P(b + 3), 256, 1, 0, 0,
                                      Kb, 256, 1, 0, 0, nullptr, 0, 0, 0, 0,
                                      nullptr, 0, 0, 0, 0, 512, 256, 256, 1, 1.0f, 0);
    gemm_wmma<<<gq, blk, 0, stream>>>(Hb, 256, 1, 0, 0, P(b + 4), 256, 1, 0, 0,
                                      Vb, 256, 1, 0, 0, nullptr, 0, 0, 0, 0,
                                      nullptr, 0, 0, 0, 0, 512, 256, 256, 1, 1.0f, 0);
    // scores = (Q/sqrt(dk)) @ K^T + bias, per (n,h): z = n*4 + h
    {
      dim3 gs(4, 4, 8);
      gemm_wmma<<<gs, blk, 0, stream>>>(
          Qb, 256, 1, 64, 65536,
          Kb, 1, 256, 64, 65536,
          Sb, 256, 1, 65536, 262144,
          Bx, 256, 1, 65536, 262144,
          nullptr, 0, 0, 0, 0,
          256, 256, 64, 4, 0.125f, 0);
    }
    softmax_k<<<dim3(2048), dim3(256), 0, stream>>>(Sb);
    // out = P @ V, scattered into (N*M, D) head-interleaved layout
    {
      dim3 ga(1, 4, 8);
      gemm_wmma<<<ga, blk, 0, stream>>>(
          Sb, 256, 1, 65536, 262144,
          Vb, 256, 1, 64, 65536,
          Ao, 256, 1, 64, 65536,
          nullptr, 0, 0, 0, 0, nullptr, 0, 0, 0, 0,
          256, 64, 256, 4, 1.0f, 0);
    }
    // x = x + out @ wo
    gemm_wmma<<<gq, blk, 0, stream>>>(Ao, 256, 1, 0, 0, P(b + 5), 256, 1, 0, 0,
                                      Xb, 256, 1, 0, 0,
                                      Xb, 256, 1, 0, 0,
                                      nullptr, 0, 0, 0, 0, 512, 256, 256, 1, 1.0f, 0);
    // FFN
    layernorm_k<<<dim3(512), dim3(256), 0, stream>>>(Xb, Hb, P(b + 6), P(b + 7));
    {
      dim3 gf(16, 8, 1);
      gemm_wmma<<<gf, blk, 0, stream>>>(Hb, 256, 1, 0, 0, P(b + 8), 1024, 1, 0, 0,
                                        Fb, 1024, 1, 0, 0,
                                        P(b + 9), 0, 1, 0, 0,
                                        nullptr, 0, 0, 0, 0,
                                        512, 1024, 256, 1, 1.0f, 1);
    }
    gemm_wmma<<<gq, blk, 0, stream>>>(Fb, 1024, 1, 0, 0, P(b + 10), 256, 1, 0, 0,
                                      Xb, 256, 1, 0, 0,
                                      P(b + 11), 0, 1, 0, 0,
                                      Xb, 256, 1, 0, 0,
                                      512, 256, 1024, 1, 1.0f, 0);
  }

  // ---------------- final norm + head ----------------
  layernorm_k<<<dim3(512), dim3(256), 0, stream>>>(Xb, Hb, P(90), P(91));
  head_k<<<dim3(2), dim3(64), 0, stream>>>(Hb, P(92), P(93), P(94), P(95),
                                           (float*)d_out);
}